// LayerDepthQKVReprojAttention_33621003993526
// MI455X (gfx1250) — compile-verified
//
#include <hip/hip_runtime.h>
#include <hip/hip_bf16.h>
#include <math.h>

// ---------------------------------------------------------------------------
// LayerDepthQKVReprojAttention for MI455X (gfx1250, wave32, WMMA).
// B=2, S=1024, D=1024, H=16, hd=64, L=4, M=12.
// All GEMMs use V_WMMA_F32_16X16X4_F32 (exact fp32) with double-buffered
// GLOBAL_LOAD_ASYNC_TO_LDS_B128 tile staging (ASYNCcnt-tracked DMA overlapping
// the WMMA math). Attention is flash-style (online softmax) with WMMA for
// Q.K^T and P.V; softmax stats live in the WMMA C-fragment row distribution
// (comp r <-> rows r / r+8) so rescaling needs no cross-lane traffic.
// ---------------------------------------------------------------------------

typedef __attribute__((ext_vector_type(2))) float v2f;
typedef __attribute__((ext_vector_type(8))) float v8f;

static constexpr size_t OUT_K = 2097152;   // d_out offset of k
static constexpr size_t OUT_V = 4194304;   // d_out offset of v
static constexpr size_t OUT_Q = 6291456;   // d_out offset of q

__device__ __forceinline__ v8f wmma4(v2f a, v2f b, v8f c) {
  // D = A(16x4 f32) * B(4x16 f32) + C(16x16 f32)
  return __builtin_amdgcn_wmma_f32_16x16x4_f32(
      false, a, false, b, (short)0, c, false, false);
}

// Async DMA: 16 bytes per lane, global -> LDS, tracked by ASYNCcnt.
__device__ __forceinline__ void async_load_b128(void* lds, const void* gmem) {
  auto l3 = (__attribute__((address_space(3))) void*)lds;
  auto g1 = (const __attribute__((address_space(1))) void*)gmem;
  asm volatile("global_load_async_to_lds_b128 %0, %1, off"
               :: "v"(l3), "v"(g1) : "memory");
}
__device__ __forceinline__ void wait_async() {
  asm volatile("s_wait_asynccnt 0" ::: "memory");
}

// mem row rid = (b*S + s)*12 + m ; source element (h,d) of that row.
__device__ __forceinline__ const float* mem_row_base(
    int rid, const float* pk, const float* pv, const float* pq) {
  int m = rid % 12;
  int bs = rid / 12;
  int s = bs & 1023, b = bs >> 10;
  const float* src;
  int l;
  if (m < 4)      { src = pq; l = m;     }
  else if (m < 8) { src = pk; l = m - 4; }
  else            { src = pv; l = m - 8; }
  // past[l,b,h,s,d] -> base + h*65536 + d
  return src + (size_t)(l * 2 + b) * 1048576 + (size_t)s * 64;
}

// ---------------------------------------------------------------------------
// Kernel 1: per-row LayerNorm stats of mem (24576 rows of 1024).
// One wave per row; 8 waves per block.
// ---------------------------------------------------------------------------
__global__ __launch_bounds__(256) void ln_stats_kernel(
    const float* __restrict__ pk, const float* __restrict__ pv,
    const float* __restrict__ pq, float* __restrict__ mu,
    float* __restrict__ rstd) {
  const int wave = threadIdx.x >> 5, lane = threadIdx.x & 31;
  const int rid = blockIdx.x * 8 + wave;
  const float* base = mem_row_base(rid, pk, pv, pq);
  float sum = 0.f, sq = 0.f;
  for (int e = 0; e < 32; ++e) {
    int idx = lane + e * 32;
    int hh = idx >> 6, dd = idx & 63;
    float xv = base[(size_t)hh * 65536 + dd];
    sum += xv;
    sq += xv * xv;
  }
  for (int off = 16; off; off >>= 1) {
    sum += __shfl_xor(sum, off, 32);
    sq  += __shfl_xor(sq,  off, 32);
  }
  if (lane == 0) {
    float mean = sum * (1.f / 1024.f);
    float var  = sq  * (1.f / 1024.f) - mean * mean;
    mu[rid]   = mean;
    rstd[rid] = rsqrtf(var + 1e-5f);
  }
}

// ---------------------------------------------------------------------------
// Kernel 2: GEMM  C = A @ W^T + bias   (row-major A: MxK, W: NxK)
// Block tile 128x64 (8 waves, each 32x32 = 2x2 WMMA tiles), BK=16.
// Double-buffered LDS staging: W (and A when ALOAD==0) via async-to-LDS DMA;
// next tile's copy overlaps current tile's 16 WMMAs.
// ALOAD: 0 = plain A (async), 1 = gather mem rows + LayerNorm (VALU path).
// EPI:   0 = row-major store MxN, 1 = qkv scatter into d_out (B,H,S,hd) parts.
// ---------------------------------------------------------------------------
template <int ALOAD, int EPI>
__global__ __launch_bounds__(256) void gemm_xwT(
    const float* __restrict__ A, const float* __restrict__ W,
    const float* __restrict__ bias, float* __restrict__ Cout,
    const float* __restrict__ pk, const float* __restrict__ pv,
    const float* __restrict__ pq, const float* __restrict__ mu,
    const float* __restrict__ rstd, int M, int N, int K) {
  constexpr int TM = 128, TN = 64, BK = 16, LDP = BK + 4;  // 20-float stride:
  // rows stay 16B-aligned for B128 LDS writes; 20*l mod 64 injective over
  // l=0..15 -> conflict-free fragment-column reads.
  __shared__ float As[2][TM][LDP];
  __shared__ float Ws[2][TN][LDP];

  const int tid = threadIdx.x;
  const int lane = tid & 31;
  const int wave = tid >> 5;
  const int l15 = lane & 15;
  const int koff = (lane >> 4) * 2;
  const int bm = blockIdx.x * TM;
  const int bn = blockIdx.y * TN;
  const int wm = (wave >> 1) * 32;  // 0,32,64,96
  const int wn = (wave & 1) * 32;   // 0,32

  auto stage = [&](int buf, int k0) {
    // ---- W tile (64 x 16): one 16B chunk per thread, async DMA ----
    {
      int r = tid >> 2, c4 = (tid & 3) << 2;
      async_load_b128(&Ws[buf][r][c4], W + (size_t)(bn + r) * K + k0 + c4);
    }
    // ---- A tile (128 x 16) ----
    if (ALOAD == 0) {
      for (int f = tid; f < TM * 4; f += 256) {
        int r = f >> 2, c4 = (f & 3) << 2;
        async_load_b128(&As[buf][r][c4], A + (size_t)(bm + r) * K + k0 + c4);
      }
    } else {
      for (int f = tid; f < TM * 4; f += 256) {
        int r = f >> 2, c4 = (f & 3) << 2;
        int rid = bm + r;
        const float* base = mem_row_base(rid, pk, pv, pq);
        int col = k0 + c4;
        const float4 tv =
            *(const float4*)(base + (size_t)(col >> 6) * 65536 + (col & 63));
        float mv = mu[rid], rs = rstd[rid];
        As[buf][r][c4]     = (tv.x - mv) * rs;
        As[buf][r][c4 + 1] = (tv.y - mv) * rs;
        As[buf][r][c4 + 2] = (tv.z - mv) * rs;
        As[buf][r][c4 + 3] = (tv.w - mv) * rs;
      }
    }
  };

  v8f acc[2][2] = {};
  stage(0, 0);
  for (int k0 = 0; k0 < K; k0 += BK) {
    const int buf = (k0 / BK) & 1;
    wait_async();       // own async copies into `buf` are done
    __syncthreads();    // everyone's copies/stores visible; prev reads retired
    if (k0 + BK < K) stage(buf ^ 1, k0 + BK);  // overlaps with math below
    for (int kk = 0; kk < BK; kk += 4) {
      v2f a0, a1, b0, b1;
      a0.x = As[buf][wm + l15][kk + koff];
      a0.y = As[buf][wm + l15][kk + koff + 1];
      a1.x = As[buf][wm + 16 + l15][kk + koff];
      a1.y = As[buf][wm + 16 + l15][kk + koff + 1];
      b0.x = Ws[buf][wn + l15][kk + koff];
      b0.y = Ws[buf][wn + l15][kk + koff + 1];
      b1.x = Ws[buf][wn + 16 + l15][kk + koff];
      b1.y = Ws[buf][wn + 16 + l15][kk + koff + 1];
      acc[0][0] = wmma4(a0, b0, acc[0][0]);
      acc[0][1] = wmma4(a0, b1, acc[0][1]);
      acc[1][0] = wmma4(a1, b0, acc[1][0]);
      acc[1][1] = wmma4(a1, b1, acc[1][1]);
    }
  }

  // ---- epilogue ----
  const int hiOff = (lane >> 4) * 8;
  for (int ti = 0; ti < 2; ++ti) {
    for (int tj = 0; tj < 2; ++tj) {
      int col = bn + wn + tj * 16 + l15;
      float bval = bias[col];
      for (int r = 0; r < 8; ++r) {
        int row = bm + wm + ti * 16 + r + hiOff;
        float val = acc[ti][tj][r] + bval;
        if (EPI == 0) {
          Cout[(size_t)row * N + col] = val;
        } else {
          // row = b*S+s ; col in [0,3072): which third -> q/k/v section
          int which = col >> 10;
          int jj = col & 1023;
          int hh = jj >> 6, dd = jj & 63;
          int b = row >> 10, s = row & 1023;
          size_t off = (which == 0) ? OUT_Q : (which == 1 ? OUT_K : OUT_V);
          Cout[off + ((size_t)(b * 16 + hh) * 1024 + s) * 64 + dd] = val;
        }
      }
    }
  }
}

// ---------------------------------------------------------------------------
// Kernel 3: flash attention + mem attention, one wave per 16 query rows.
// grid = B*H*(S/64) blocks of 128 threads (4 independent waves).
// Softmax stats m,l kept per C-fragment component (comp r <-> row r / r+8).
// ---------------------------------------------------------------------------
__global__ __launch_bounds__(128) void attn_kernel(
    const float* __restrict__ q, const float* __restrict__ k,
    const float* __restrict__ v, const float* __restrict__ rk,
    const float* __restrict__ rv, float* __restrict__ merged) {
  const int wave = threadIdx.x >> 5, lane = threadIdx.x & 31;
  const int l15 = lane & 15, hi = lane >> 4;
  const int koff = hi * 2;
  const int bid = blockIdx.x;
  const int sb = bid & 15;   // S/64 chunks
  const int bh = bid >> 4;   // b*16 + h
  const int h = bh & 15, b = bh >> 4;
  const int s0 = sb * 64 + wave * 16;

  const float* Q  = q + (size_t)bh * 65536;
  const float* Km = k + (size_t)bh * 65536;
  const float* Vm = v + (size_t)bh * 65536;

  __shared__ float Pl[4][16][17];   // per-wave P-tile staging (C -> A layout)
  __shared__ float Sm[4][16][12];   // per-wave mem scores

  const float scale = 0.125f;  // 1/sqrt(64)

  // preload Q fragments (scale folded in)
  v2f qf[16];
  for (int j = 0; j < 16; ++j) {
    const float* qp = Q + (size_t)(s0 + l15) * 64 + 4 * j + koff;
    qf[j].x = qp[0] * scale;
    qf[j].y = qp[1] * scale;
  }

  v8f ctx[4] = {};            // 16x64 ctx as 4 d-tiles
  float mrow[8], lrow[8];
  for (int r = 0; r < 8; ++r) { mrow[r] = -INFINITY; lrow[r] = 0.f; }

  for (int t0 = 0; t0 <= s0; t0 += 16) {
    // scores tile = Q(16x64) . K^T(64x16)
    v8f sc = {};
    for (int j = 0; j < 16; ++j) {
      v2f bf;
      const float* kp = Km + (size_t)(t0 + l15) * 64 + 4 * j + koff;
      bf.x = kp[0];
      bf.y = kp[1];
      sc = wmma4(qf[j], bf, sc);
    }
    if (t0 == s0) {  // causal mask on diagonal block
      for (int r = 0; r < 8; ++r) {
        int row = r + hi * 8;
        if (l15 > row) sc[r] = -INFINITY;
      }
    }
    // online softmax update per row (comp r)
    for (int r = 0; r < 8; ++r) {
      float mx = sc[r];
      for (int off = 8; off; off >>= 1) mx = fmaxf(mx, __shfl_xor(mx, off, 32));
      float mnew = fmaxf(mrow[r], mx);
      float alpha = __expf(mrow[r] - mnew);
      mrow[r] = mnew;
      float p = __expf(sc[r] - mnew);
      float rs = p;
      for (int off = 8; off; off >>= 1) rs += __shfl_xor(rs, off, 32);
      lrow[r] = lrow[r] * alpha + rs;
      for (int t = 0; t < 4; ++t) ctx[t][r] *= alpha;
      Pl[wave][r + hi * 8][l15] = p;     // C layout -> LDS
    }
    asm volatile("s_wait_dscnt 0" ::: "memory");  // wave-local LDS transpose
    // ctx += P(16x16) . V(16x64)
    for (int j = 0; j < 4; ++j) {
      v2f af;
      af.x = Pl[wave][l15][4 * j + koff];
      af.y = Pl[wave][l15][4 * j + koff + 1];
      for (int t = 0; t < 4; ++t) {
        v2f bf;
        const float* vp = Vm + (size_t)(t0 + 4 * j + koff) * 64 + t * 16 + l15;
        bf.x = vp[0];
        bf.y = vp[64];
        ctx[t] = wmma4(af, bf, ctx[t]);
      }
    }
  }

  // ---- 12 mem scores per row (scalar; tiny) ----
  for (int idx = lane; idx < 192; idx += 32) {
    int row = idx / 12, m = idx % 12;
    const float* qp = Q + (size_t)(s0 + row) * 64;
    const float* rp =
        rk + ((size_t)((b * 1024 + s0 + row) * 12 + m)) * 1024 + h * 64;
    float d = 0.f;
    for (int e = 0; e < 64; ++e) d += qp[e] * rp[e];
    Sm[wave][row][m] = d * scale;
  }
  asm volatile("s_wait_dscnt 0" ::: "memory");

  for (int r = 0; r < 8; ++r) {
    int row = r + hi * 8;
    float smv[12];
    float mx = mrow[r];
    for (int m = 0; m < 12; ++m) {
      smv[m] = Sm[wave][row][m];
      mx = fmaxf(mx, smv[m]);
    }
    float alpha = __expf(mrow[r] - mx);
    float pm[12];
    float rs = 0.f;
    for (int m = 0; m < 12; ++m) {
      pm[m] = __expf(smv[m] - mx);
      rs += pm[m];
    }
    lrow[r] = lrow[r] * alpha + rs;
    const float* rvb =
        rv + ((size_t)(b * 1024 + s0 + row)) * 12288 + h * 64;
    for (int t = 0; t < 4; ++t) {
      float add = 0.f;
      const float* rp = rvb + t * 16 + l15;
      for (int m = 0; m < 12; ++m) add += pm[m] * rp[(size_t)m * 1024];
      ctx[t][r] = ctx[t][r] * alpha + add;
    }
    float inv = 1.f / lrow[r];
    for (int t = 0; t < 4; ++t) ctx[t][r] *= inv;
  }

  // store merged[b, s, h*64 + d]
  for (int r = 0; r < 8; ++r) {
    int row = r + hi * 8;
    float* mp = merged + ((size_t)(b * 1024 + s0 + row)) * 1024 + h * 64;
    for (int t = 0; t < 4; ++t) mp[t * 16 + l15] = ctx[t][r];
  }
}

// ---------------------------------------------------------------------------
extern "C" void kernel_launch(void* const* d_in, const int* in_sizes, int n_in,
                              void* d_out, int out_size, void* d_ws,
                              size_t ws_size, hipStream_t stream) {
  const float* x      = (const float*)d_in[0];
  const float* past_k = (const float*)d_in[1];
  const float* past_v = (const float*)d_in[2];
  const float* past_q = (const float*)d_in[3];
  const float* Wqkv   = (const float*)d_in[4];
  const float* bqkv   = (const float*)d_in[5];
  const float* Wout   = (const float*)d_in[6];
  const float* bout   = (const float*)d_in[7];
  float* out = (float*)d_out;
  float* ws  = (float*)d_ws;

  float* rk     = ws;                 // 24576*1024
  float* rv     = ws + 25165824;      // 24576*1024
  float* merged = ws + 50331648;      // 2048*1024
  float* mu     = ws + 52428800;      // 24576
  float* rstd   = ws + 52453376;      // 24576

  // 1. LayerNorm stats of mem rows (fused normalize happens in rk/rv GEMMs)
  ln_stats_kernel<<<3072, 256, 0, stream>>>(past_k, past_v, past_q, mu, rstd);

  // 2. qkv projection -> q/k/v sections of d_out in (B,H,S,hd) layout
  {
    dim3 g(2048 / 128, 3072 / 64);
    gemm_xwT<0, 1><<<g, 256, 0, stream>>>(x, Wqkv, bqkv, out, nullptr, nullptr,
                                          nullptr, nullptr, nullptr, 2048,
                                          3072, 1024);
  }

  // 3/4. rk = LN(mem) @ Wk^T + bk ; rv = LN(mem) @ Wv^T + bv
  {
    dim3 g(24576 / 128, 1024 / 64);
    gemm_xwT<1, 0><<<g, 256, 0, stream>>>(nullptr, Wqkv + 1048576, bqkv + 1024,
                                          rk, past_k, past_v, past_q, mu, rstd,
                                          24576, 1024, 1024);
    gemm_xwT<1, 0><<<g, 256, 0, stream>>>(nullptr, Wqkv + 2097152, bqkv + 2048,
                                          rv, past_k, past_v, past_q, mu, rstd,
                                          24576, 1024, 1024);
  }

  // 5. flash attention (token + mem) -> merged (B,S,D)
  attn_kernel<<<512, 128, 0, stream>>>(out + OUT_Q, out + OUT_K, out + OUT_V,
                                       rk, rv, merged);

  // 6. out = merged @ Wout^T + bout
  {
    dim3 g(2048 / 128, 1024 / 64);
    gemm_xwT<0, 0><<<g, 256, 0, stream>>>(merged, Wout, bout, out, nullptr,
                                          nullptr, nullptr, nullptr, nullptr,
                                          2048, 1024, 1024);
  }
}